// MultiHeadAttentionLayer_2628519985867
// MI455X (gfx1250) — compile-verified
//
#include <hip/hip_runtime.h>
#include <hip/hip_bf16.h>

#define B 2
#define S 512
#define HID 512
#define NH 8
#define HD 64

typedef __attribute__((ext_vector_type(2))) float v2f;
typedef __attribute__((ext_vector_type(8))) float v8f;

// ---------------------------------------------------------------------------
// native tanh (gfx1250 V_TANH_F32, TRANS32 op that co-executes with VALU)
// ---------------------------------------------------------------------------
__device__ __forceinline__ float dtanh(float x) {
#if __has_builtin(__builtin_amdgcn_tanhf)
    return __builtin_amdgcn_tanhf(x);
#else
    float y;
    asm("v_tanh_f32 %0, %1" : "=v"(y) : "v"(x));
    return y;
#endif
}

// ---------------------------------------------------------------------------
// Generic GEMM with transposed weight:  D[m,n] = sum_k A[m,k]*W[n,k] (+ bias[n])
// One wave computes one 16x16 tile of D via V_WMMA_F32_16X16X4_F32.
// A frag: lane(0..15) holds A[m=l16][k0..k0+1], lane(16..31) A[m=l16][k0+2..3]
// B frag: lane(0..15) holds W[n=l16][k0..k0+1], lane(16..31) W[n=l16][k0+2..3]
// Both fragments are contiguous 8-byte loads.
// C/D: lane<16 -> rows 0..7 of col l16; lane>=16 -> rows 8..15 of col l16.
// Grid: x = M/16 tiles, (y*blockDim.y+ty) = N/16 tiles.  blockDim = (32,4).
// ---------------------------------------------------------------------------
__global__ __launch_bounds__(128) void proj_gemm_wmma(
    const float* __restrict__ A, const float* __restrict__ W,
    const float* __restrict__ bias, float* __restrict__ D,
    int K, int lda, int ldw, int ldd)
{
    const int mbase = blockIdx.x * 16;
    const int nbase = (blockIdx.y * blockDim.y + threadIdx.y) * 16;
    const int lane  = threadIdx.x & 31;
    const int l16   = lane & 15;
    const int half  = lane >> 4;

    const float* Ar = A + (size_t)(mbase + l16) * lda;
    const float* Wr = W + (size_t)(nbase + l16) * ldw;

    v8f acc = {};
#pragma unroll 8
    for (int k0 = 0; k0 < K; k0 += 4) {
        v2f af = *(const v2f*)&Ar[k0 + 2 * half];
        v2f bf = *(const v2f*)&Wr[k0 + 2 * half];
        acc = __builtin_amdgcn_wmma_f32_16x16x4_f32(
            false, af, false, bf, (short)0, acc, false, false);
    }

    const float bb = bias ? bias[nbase + l16] : 0.0f;
#pragma unroll
    for (int r = 0; r < 8; ++r)
        D[(size_t)(mbase + half * 8 + r) * ldd + nbase + l16] = acc[r] + bb;
}

// ---------------------------------------------------------------------------
// Fused additive-attention energy + mask + softmax.
//   energy[b,h,q,k] = Vb + sum_d Vw[d]*tanh(Qt[b,q,h,d] + Kt[b,k,h,d])
//   attn = softmax_k(where(mask==0, -1e10, energy))   -> d_out attention region
// Qt/Kt stored in [B,S,NH,HD] flat layout (row stride NH*HD = 512 per s).
// Block = 256 threads = 8 waves; wave w owns q-row q0+w; lane covers 16 k's
// (2 columns per k-tile).  Kt tiled 64x64 into LDS with pitch 66 so the
// ds_load_b64 column reads are bank-conflict-free (lane i -> banks {2i,2i+1}).
// Inner loop: 4x ds_load_b64 per 4 tanh + 8 fma/add; qt/vw broadcasts are
// amortized over both columns.  Next tile prefetched via global_prefetch_b8.
// ---------------------------------------------------------------------------
#define EROWS 8
#define KTILE 64

__global__ __launch_bounds__(256) void energy_softmax(
    const float* __restrict__ Qt, const float* __restrict__ Kt,
    const float* __restrict__ Vw, const float* __restrict__ Vb,
    const int* __restrict__ mask, float* __restrict__ attn)
{
    const int bh = blockIdx.y;
    const int b  = bh >> 3;
    const int h  = bh & 7;
    const int q0 = blockIdx.x * EROWS;
    const int wave = threadIdx.x >> 5;
    const int lane = threadIdx.x & 31;

    __shared__ float sKt[KTILE][66];
    __shared__ float sQt[EROWS][64];
    __shared__ float sVw[64];

    if (threadIdx.x < 64) sVw[threadIdx.x] = Vw[threadIdx.x];
    for (int i = threadIdx.x; i < EROWS * 64; i += 256) {
        int r = i >> 6, d = i & 63;
        sQt[r][d] = Qt[(((size_t)b * S + q0 + r) * NH + h) * HD + d];
    }

    const float vb = Vb[0];
    float ev[16];

#pragma unroll 1
    for (int t = 0; t < S / KTILE; ++t) {
        __syncthreads();
        // stage Kt tile: 64 rows x 64 d = 1024 float4s across 256 threads
        for (int i = threadIdx.x; i < KTILE * 64 / 4; i += 256) {
            int r = i >> 4, d4 = (i & 15) * 4;
            const float4 v = *(const float4*)&Kt[(((size_t)b * S + t * KTILE + r) * NH + h) * HD + d4];
            sKt[r][d4 + 0] = v.x; sKt[r][d4 + 1] = v.y;
            sKt[r][d4 + 2] = v.z; sKt[r][d4 + 3] = v.w;
        }
        __syncthreads();

        // warm the cache for the next tile (global_prefetch_b8, no counters)
        if (t + 1 < S / KTILE && threadIdx.x < 128) {
            const int r = threadIdx.x >> 1, o = (threadIdx.x & 1) * 32;
            __builtin_prefetch(
                &Kt[(((size_t)b * S + (t + 1) * KTILE + r) * NH + h) * HD + o], 0, 0);
        }

        // two k-columns per lane, both driven by one pass over d
        const int c0 = lane, c1 = 32 + lane;
        float a0 = vb, a1 = vb;
#pragma unroll
        for (int d = 0; d < 64; d += 2) {
            const float2 q  = *(const float2*)&sQt[wave][d];
            const float2 w  = *(const float2*)&sVw[d];
            const float2 x0 = *(const float2*)&sKt[c0][d];
            const float2 x1 = *(const float2*)&sKt[c1][d];
            a0 += w.x * dtanh(q.x + x0.x);
            a1 += w.x * dtanh(q.x + x1.x);
            a0 += w.y * dtanh(q.y + x0.y);
            a1 += w.y * dtanh(q.y + x1.y);
        }
        ev[t * 2 + 0] = a0;
        ev[t * 2 + 1] = a1;
    }

    // mask + softmax over the 512 columns of this q-row (16 vals/lane, wave32)
    float m = -3.0e38f;
#pragma unroll
    for (int i = 0; i < 16; ++i) {
        const int k = (i >> 1) * 64 + (i & 1) * 32 + lane;
        if (mask[b * S + k] == 0) ev[i] = -1.0e10f;
        m = fmaxf(m, ev[i]);
    }
#pragma unroll
    for (int off = 16; off > 0; off >>= 1) m = fmaxf(m, __shfl_xor(m, off, 32));
    float ssum = 0.0f;
#pragma unroll
    for (int i = 0; i < 16; ++i) { ev[i] = __expf(ev[i] - m); ssum += ev[i]; }
#pragma unroll
    for (int off = 16; off > 0; off >>= 1) ssum += __shfl_xor(ssum, off, 32);
    const float inv = 1.0f / ssum;

    const size_t base = ((size_t)bh * S + (q0 + wave)) * S;
#pragma unroll
    for (int i = 0; i < 16; ++i) {
        const int k = (i >> 1) * 64 + (i & 1) * 32 + lane;
        attn[base + k] = ev[i] * inv;
    }
}

// ---------------------------------------------------------------------------
// Batched attn @ V, one wave per 16x16 tile, writing directly in [B,S,HID]
// layout so the Wo projection needs no transpose.
//   Xh[b, q, h*64+d] = sum_k attn[b,h,q,k] * V[b,k,h*64+d]
// Grid: (S/16, 1, B*NH), block (32,4): ty covers the 4 N-tiles of HD=64.
// ---------------------------------------------------------------------------
__global__ __launch_bounds__(128) void attnV_gemm_wmma(
    const float* __restrict__ attn, const float* __restrict__ Vp,
    float* __restrict__ Xh)
{
    const int bh = blockIdx.z;
    const int b  = bh >> 3;
    const int h  = bh & 7;
    const int mbase = blockIdx.x * 16;
    const int nbase = threadIdx.y * 16;
    const int lane  = threadIdx.x & 31;
    const int l16   = lane & 15;
    const int half  = lane >> 4;

    const float* A  = attn + (size_t)bh * S * S;                 // lda = S
    const float* Bm = Vp + ((size_t)b * S * NH + h) * HD;        // B[k][n] = Bm[k*512 + n]
    float*       D  = Xh + ((size_t)b * S * NH + h) * HD;        // D[m][n] = D[m*512 + n]

    v8f acc = {};
#pragma unroll 8
    for (int k0 = 0; k0 < S; k0 += 4) {
        v2f af = *(const v2f*)&A[(size_t)(mbase + l16) * S + k0 + 2 * half];
        v2f bf;
        bf.x = Bm[(size_t)(k0 + 2 * half) * HID + nbase + l16];
        bf.y = Bm[(size_t)(k0 + 2 * half + 1) * HID + nbase + l16];
        acc = __builtin_amdgcn_wmma_f32_16x16x4_f32(
            false, af, false, bf, (short)0, acc, false, false);
    }
#pragma unroll
    for (int r = 0; r < 8; ++r)
        D[(size_t)(mbase + half * 8 + r) * HID + nbase + l16] = acc[r];
}

// ---------------------------------------------------------------------------
// Host-side orchestration
// ---------------------------------------------------------------------------
extern "C" void kernel_launch(void* const* d_in, const int* in_sizes, int n_in,
                              void* d_out, int out_size, void* d_ws, size_t ws_size,
                              hipStream_t stream) {
    const float* query = (const float*)d_in[0];
    const float* key_  = (const float*)d_in[1];
    const float* value = (const float*)d_in[2];
    const int*   mask  = (const int*)  d_in[3];
    const float* Wq = (const float*)d_in[4];  const float* bq = (const float*)d_in[5];
    const float* Wk = (const float*)d_in[6];  const float* bk = (const float*)d_in[7];
    const float* Wv = (const float*)d_in[8];  const float* bv = (const float*)d_in[9];
    const float* Wo = (const float*)d_in[10]; const float* bo = (const float*)d_in[11];
    const float* Waw = (const float*)d_in[12]; const float* Wab = (const float*)d_in[13];
    const float* Uaw = (const float*)d_in[14]; const float* Uab = (const float*)d_in[15];
    const float* Vw  = (const float*)d_in[16]; const float* Vb  = (const float*)d_in[17];

    const size_t NTOK = (size_t)B * S;          // 1024 rows of HID
    float* ws = (float*)d_ws;
    float* Qp  = ws;                  // [B*S, HID]
    float* Kp  = Qp  + NTOK * HID;
    float* Vp  = Kp  + NTOK * HID;
    float* Qt  = Vp  + NTOK * HID;    // [B*S*NH, HD] (== [B,S,NH,HD] flat)
    float* Kt  = Qt  + NTOK * HID;
    float* Xh  = Kt  + NTOK * HID;    // attn@V result, already [B,S,HID]

    float* x_out    = (float*)d_out;                 // [B,S,HID]
    float* attn_out = x_out + NTOK * HID;            // [B,NH,S,S]

    dim3 blkG(32, 4);

    // Q/K/V projections: M=1024, N=512, K=512
    {
        dim3 grd(NTOK / 16, HID / 64);
        proj_gemm_wmma<<<grd, blkG, 0, stream>>>(query, Wq, bq, Qp, HID, HID, HID, HID);
        proj_gemm_wmma<<<grd, blkG, 0, stream>>>(key_,  Wk, bk, Kp, HID, HID, HID, HID);
        proj_gemm_wmma<<<grd, blkG, 0, stream>>>(value, Wv, bv, Vp, HID, HID, HID, HID);
    }

    // Per-head Qt/Kt: flat [8192,64] x [64,64]^T
    {
        dim3 grd(NTOK * NH / 16, 1);
        proj_gemm_wmma<<<grd, blkG, 0, stream>>>(Qp, Waw, Wab, Qt, HD, HD, HD, HD);
        proj_gemm_wmma<<<grd, blkG, 0, stream>>>(Kp, Uaw, Uab, Kt, HD, HD, HD, HD);
    }

    // Fused energy + tanh + mask + softmax -> attention output
    {
        dim3 grd(S / EROWS, B * NH);
        energy_softmax<<<grd, dim3(256), 0, stream>>>(Qt, Kt, Vw, Vb, mask, attn_out);
    }

    // attn @ V (batched over B*NH), written straight into [B,S,HID]
    {
        dim3 grd(S / 16, 1, B * NH);
        attnV_gemm_wmma<<<grd, blkG, 0, stream>>>(attn_out, Vp, Xh);
    }

    // Output projection: x = Xh @ Wo^T + bo
    {
        dim3 grd(NTOK / 16, HID / 64);
        proj_gemm_wmma<<<grd, blkG, 0, stream>>>(Xh, Wo, bo, x_out, HID, HID, HID, HID);
    }
}